// Feat2Dist_batch_14499809591877
// MI455X (gfx1250) — compile-verified
//
#include <hip/hip_runtime.h>
#include <hip/hip_bf16.h>

// Feat2Dist: y[b,i,j] = ||x_i||^2 + ||x_j||^2 - 2 * <x_i, x_j>
// Input  [8, 256, 64, 64] fp32  == per-batch X^T [K=256][N=4096], row-major
// Output [8, 4096, 4096] fp32
//
// Compute-bound (68.7 GFLOP vs ~24us HBM time at 23.3 TB/s) -> Gram via
// V_WMMA_F32_16X16X4_F32 (exact fp32, matches reference numerics).
// 256x128 block tile, 8 waves each owning a 64x64 register tile
// (16 v8f accumulators) -> 1.0 LDS dword-read per WMMA instead of 1.5.

typedef __attribute__((ext_vector_type(2))) float v2f;
typedef __attribute__((ext_vector_type(8))) float v8f;

namespace {
constexpr int D     = 256;         // feature dim (GEMM K)
constexpr int N     = 4096;        // 64*64 points
constexpr int TM    = 256;         // block tile rows
constexpr int TN    = 128;         // block tile cols
constexpr int KB    = 32;          // K slab staged in LDS
constexpr int LDAA  = TM + 16;     // 272: rows k,k+2 land 32 banks apart
constexpr int LDAB  = TN + 16;     // 144: same property
}

__global__ __launch_bounds__(256) void feat2dist_wmma_f32(
    const float* __restrict__ in, float* __restrict__ out) {
  __shared__ float lsA[KB * LDAA];   // X^T slab for the 256 output rows
  __shared__ float lsB[KB * LDAB];   // X^T slab for the 128 output cols
  __shared__ float sxnA[TM];
  __shared__ float sxnB[TN];

  const int b  = blockIdx.z;
  const int i0 = blockIdx.y * TM;             // output row base
  const int j0 = blockIdx.x * TN;             // output col base
  const float* inb = in + (size_t)b * D * N;  // X^T for this batch

  const int t    = threadIdx.x;   // 0..255 (8 waves, wave32)
  const int lane = t & 31;
  const int wave = t >> 5;
  const int wrow = wave >> 1;     // 0..3 : 64-row band
  const int wcol = wave & 1;      // 0..1 : 64-col band
  const int half = lane >> 4;     // K-half select for f32 WMMA operand layout
  const int l    = lane & 15;     // M/N index within fragment

  v8f acc[4][4] = {};             // 16 x 16x16 fp32 accumulators per wave
  float xnA = 0.f, xnB = 0.f;     // per-thread column-norm partials

  for (int k0 = 0; k0 < D; k0 += KB) {
    __syncthreads();
    // ---- stage K-slab, b128 coalesced ----
#pragma unroll
    for (int i = 0; i < 8; ++i) {               // A: 32 rows x 256 floats
      const int idx = t + i * 256;              // 0..2047
      const int kr  = idx >> 6;                 // 0..31
      const int c4  = (idx & 63) << 2;          // 0..252
      const float4 a4 = *(const float4*)(inb + (size_t)(k0 + kr) * N + i0 + c4);
      *(float4*)(&lsA[kr * LDAA + c4]) = a4;
    }
#pragma unroll
    for (int i = 0; i < 4; ++i) {               // B: 32 rows x 128 floats
      const int idx = t + i * 256;              // 0..1023
      const int kr  = idx >> 5;                 // 0..31
      const int c4  = (idx & 31) << 2;          // 0..124
      const float4 b4 = *(const float4*)(inb + (size_t)(k0 + kr) * N + j0 + c4);
      *(float4*)(&lsB[kr * LDAB + c4]) = b4;
    }
    __syncthreads();

    // ---- squared-norm partials straight from the staged slabs ----
    {
      float s = 0.f;
#pragma unroll
      for (int k = 0; k < KB; ++k) {
        const float v = lsA[k * LDAA + t];
        s += v * v;
      }
      xnA += s;
      if (t < TN) {
        float s2 = 0.f;
#pragma unroll
        for (int k = 0; k < KB; ++k) {
          const float v = lsB[k * LDAB + t];
          s2 += v * v;
        }
        xnB += s2;
      }
    }

    // ---- WMMA over the slab: K in steps of 4, 16 WMMAs per step ----
#pragma unroll
    for (int kk = 0; kk < KB; kk += 4) {
      const int kb = kk + 2 * half;   // f32 A/B layout: half-wave -> K {0,1} vs {2,3}
      v2f aF[4], bF[4];
#pragma unroll
      for (int rb = 0; rb < 4; ++rb) {
        const int col = wrow * 64 + rb * 16 + l;    // A: M = l
        aF[rb].x = lsA[kb * LDAA + col];
        aF[rb].y = lsA[(kb + 1) * LDAA + col];
      }
#pragma unroll
      for (int cb = 0; cb < 4; ++cb) {
        const int col = wcol * 64 + cb * 16 + l;    // B: N = l
        bF[cb].x = lsB[kb * LDAB + col];
        bF[cb].y = lsB[(kb + 1) * LDAB + col];
      }
#pragma unroll
      for (int rb = 0; rb < 4; ++rb)
#pragma unroll
        for (int cb = 0; cb < 4; ++cb)
          acc[rb][cb] = __builtin_amdgcn_wmma_f32_16x16x4_f32(
              false, aF[rb], false, bF[cb], (short)0, acc[rb][cb], false, false);
    }
  }

  // ---- publish norms, then fused epilogue ----
  sxnA[t] = xnA;
  if (t < TN) sxnB[t] = xnB;
  __syncthreads();

  float* outb = out + (size_t)b * N * N;
#pragma unroll
  for (int rb = 0; rb < 4; ++rb) {
    const int ib = wrow * 64 + rb * 16 + 8 * half;   // C/D: row = v + 8*half
#pragma unroll
    for (int cb = 0; cb < 4; ++cb) {
      const int jb = wcol * 64 + cb * 16 + l;        // C/D: col = lane&15
      const float xj = sxnB[jb];
#pragma unroll
      for (int v = 0; v < 8; ++v) {
        const int i = ib + v;
        outb[(size_t)(i0 + i) * N + (j0 + jb)] = sxnA[i] + xj - 2.0f * acc[rb][cb][v];
      }
    }
  }
}

extern "C" void kernel_launch(void* const* d_in, const int* in_sizes, int n_in,
                              void* d_out, int out_size, void* d_ws, size_t ws_size,
                              hipStream_t stream) {
  (void)in_sizes; (void)n_in; (void)d_ws; (void)ws_size; (void)out_size;
  const float* in = (const float*)d_in[0];
  float* out = (float*)d_out;
  dim3 grid(N / TN, N / TM, 8);   // 32 x 16 tiles x 8 batches
  dim3 block(256);
  feat2dist_wmma_f32<<<grid, block, 0, stream>>>(in, out);
}